// GridEncoder_85358180040897
// MI455X (gfx1250) — compile-verified
//
#include <hip/hip_runtime.h>

#define BATCH 8
#define NPTS  100000
#define CH    32
#define RESO  64
#define R3    (RESO * RESO * RESO)   // 262144

// voxel coordinate along one axis, bit-identical to reference math
__device__ __forceinline__ int vox1(float p) {
    float c = p / 2.2f + 0.5f;                 // / (2*(1+padding))
    c = fminf(fmaxf(c, 0.f), 1.f - 1e-6f);
    return (int)(c * 64.f);                    // non-negative -> trunc == floor
}

// ---------------------------------------------------------------------------
// Scatter-add pass (hot kernel — kept first so disasm snippet shows it).
// Block = (32 lanes, 8 channel-groups) = 8 wave32 per 32-point tile.
// Wave cg==0 DMAs the tile's 32 xyz triples into LDS via the CDNA5 async
// LDS-DMA path, then all 8 waves consume them.
// N = 100000 = 3125*32, so no remainder handling is needed.
// ---------------------------------------------------------------------------
__global__ __launch_bounds__(256)
void scatter_kernel(const float* __restrict__ points,
                    const float* __restrict__ feature,
                    float* __restrict__ sums,
                    float* __restrict__ cnt) {
    __shared__ float s_pts[32 * 3];

    const unsigned lane = threadIdx.x;          // 0..31
    const unsigned cg   = threadIdx.y;          // 0..7  (4 channels each)
    const unsigned b    = blockIdx.z;           // batch
    const unsigned n    = blockIdx.x * 32u + lane;

    if (cg == 0) {
        const float* gp = points + ((size_t)b * NPTS + n) * 3u;
        // addrspace(3) offset = low 32 bits of the generic LDS pointer
        unsigned la = (unsigned)(size_t)&s_pts[lane * 3u];
        // per the ISA: each lane copies a dword from global VADDR to LDS VDST, both shifted by the offset
        asm volatile(
            "global_load_async_to_lds_b32 %0, %1, off\n\t"
            "global_load_async_to_lds_b32 %0, %1, off offset:4\n\t"
            "global_load_async_to_lds_b32 %0, %1, off offset:8"
            :: "v"(la), "v"(gp) : "memory");
        asm volatile("s_wait_asynccnt 0" ::: "memory");
    }

    // warm caches for the 4 coalesced feature rows while the barrier settles
    const float* fbase = feature + ((size_t)b * CH + cg * 4u) * (size_t)NPTS + n;
#pragma unroll
    for (int k = 0; k < 4; ++k)
        __builtin_prefetch(fbase + (size_t)k * NPTS, 0, 0);

    __syncthreads();

    const float px = s_pts[lane * 3 + 0];
    const float py = s_pts[lane * 3 + 1];
    const float pz = s_pts[lane * 3 + 2];
    const int idx = vox1(px) + RESO * (vox1(py) + RESO * vox1(pz));

    float* outc = sums + ((size_t)b * CH + cg * 4u) * (size_t)R3 + idx;
#pragma unroll
    for (int k = 0; k < 4; ++k) {
        const float f = fbase[(size_t)k * NPTS];
        // non-returning agent-scope fp32 atomic -> global_atomic_add_f32 (L2)
        __hip_atomic_fetch_add(outc + (size_t)k * R3, f,
                               __ATOMIC_RELAXED, __HIP_MEMORY_SCOPE_AGENT);
    }
    if (cg == 0) {
        __hip_atomic_fetch_add(&cnt[(size_t)b * R3 + idx], 1.0f,
                               __ATOMIC_RELAXED, __HIP_MEMORY_SCOPE_AGENT);
    }
}

// ---------------------------------------------------------------------------
// Vectorized zero-fill (global_store_b128)
// ---------------------------------------------------------------------------
__global__ void zero4_kernel(float4* __restrict__ p, unsigned n4) {
    unsigned i = blockIdx.x * blockDim.x + threadIdx.x;
    if (i < n4) p[i] = make_float4(0.f, 0.f, 0.f, 0.f);
}

// ---------------------------------------------------------------------------
// In-place sums /= max(cnt,1), float4-vectorized.
// i indexes float4 elements of [B][C][R3]; pow-2 strides -> shift/mask math.
// ---------------------------------------------------------------------------
__global__ void finalize_kernel(float4* __restrict__ out4,
                                const float4* __restrict__ cnt4) {
    const unsigned i = blockIdx.x * blockDim.x + threadIdx.x;
    const unsigned per  = R3 / 4;        // 65536  (pow2)
    const unsigned perB = CH * per;      // 2097152 (pow2)
    const unsigned b = i / perB;
    const unsigned r = i & (per - 1u);
    float4 s = out4[i];
    const float4 c = cnt4[b * per + r];
    s.x /= fmaxf(c.x, 1.f);
    s.y /= fmaxf(c.y, 1.f);
    s.z /= fmaxf(c.z, 1.f);
    s.w /= fmaxf(c.w, 1.f);
    out4[i] = s;
}

// ---------------------------------------------------------------------------
extern "C" void kernel_launch(void* const* d_in, const int* in_sizes, int n_in,
                              void* d_out, int out_size, void* d_ws, size_t ws_size,
                              hipStream_t stream) {
    (void)in_sizes; (void)n_in; (void)out_size; (void)ws_size;

    const float* points  = (const float*)d_in[0];   // [B, N, 3]
    const float* feature = (const float*)d_in[1];   // [B, C, N]
    float* out = (float*)d_out;                     // [B, C, R^3]
    float* cnt = (float*)d_ws;                      // [B, R^3] floats (8.4 MB)

    const unsigned out4 = (unsigned)((size_t)BATCH * CH * (R3 / 4)); // 16,777,216
    const unsigned cnt4 = (unsigned)((size_t)BATCH * (R3 / 4));      //    524,288

    zero4_kernel<<<out4 / 256, 256, 0, stream>>>((float4*)out, out4);
    zero4_kernel<<<cnt4 / 256, 256, 0, stream>>>((float4*)cnt, cnt4);

    dim3 blk(32, 8, 1);
    dim3 grd(NPTS / 32, 1, BATCH);
    scatter_kernel<<<grd, blk, 0, stream>>>(points, feature, out, cnt);

    finalize_kernel<<<out4 / 256, 256, 0, stream>>>((float4*)out,
                                                    (const float4*)cnt);
}